// RPQEmbedding_3255585210640
// MI455X (gfx1250) — compile-verified
//
#include <hip/hip_runtime.h>
#include <hip/hip_bf16.h>
#include <stdint.h>

// ---------------------------------------------------------------------------
// RPQ embedding gather for gfx1250 (MI455X).
//   out[tok, h*128 + d] = codebooks[h, indices[h, input[tok]], d]
// Zero FLOPs -> store-bandwidth bound (134 MB out @ 23.3 TB/s ~ 5.8 us).
// Strategy: block = (codebook h, 64-token tile). Stage the 128 KB codebook
// into LDS with the Tensor Data Mover (tensor_load_to_lds, TENSORcnt), then
// each wave32 gathers one 512B row from LDS (ds_load_b128) and streams it to
// global with fully coalesced global_store_b128 (32 lanes x 16B).
// ---------------------------------------------------------------------------

#define RPQ_NUM_CB   16
#define RPQ_CB_DIM   128
#define RPQ_CODES    256
#define RPQ_CB_ELEMS (RPQ_CODES * RPQ_CB_DIM)   // 32768 floats = 128 KB
#define RPQ_EMBED    (RPQ_NUM_CB * RPQ_CB_DIM)  // 2048
#define RPQ_TOK_PER_BLK 64                      // 8 waves x 8 iterations

typedef __attribute__((ext_vector_type(4))) unsigned int v4u_t;
typedef __attribute__((ext_vector_type(8))) int          v8i_t;
typedef __attribute__((ext_vector_type(4))) int          v4i_t;

__global__ void __launch_bounds__(256)
rpq_embed_kernel(const int* __restrict__ input,        // (num_tokens)
                 const int* __restrict__ indices,      // (16, num_emb)
                 const float* __restrict__ codebooks,  // (16, 256, 128)
                 float* __restrict__ out,              // (num_tokens, 2048)
                 int num_tokens, int num_emb)
{
    extern __shared__ float lds[];                     // 128 KB: codebook h
    const int h   = blockIdx.y;
    const unsigned tid = threadIdx.x;

#if __has_builtin(__builtin_amdgcn_tensor_load_to_lds) && __has_builtin(__builtin_amdgcn_s_wait_tensorcnt)
    // ---- TDM stage: one DMA of the whole 256x128 fp32 codebook into LDS ----
    if (tid < 32u) {  // wave 0 only, EXEC all-ones; TDM ignores EXEC but
                      // other waves branch around the issue entirely.
        const unsigned long long gaddr =
            (unsigned long long)(const void*)(codebooks + (size_t)h * RPQ_CB_ELEMS);
        const unsigned lds_addr = (unsigned)(unsigned long long)(const void*)lds;

        // D# group 0 (128b): count=1, lds_addr, global_addr[56:0], type=2.
        v4u_t g0;
        g0[0] = 1u;                                        // count=1, user desc
        g0[1] = lds_addr;                                  // LDS byte address
        g0[2] = (unsigned)(gaddr & 0xFFFFFFFFu);           // global_addr[31:0]
        g0[3] = (unsigned)((gaddr >> 32) & 0x01FFFFFFu)    // global_addr[56:32]
              | (2u << 30);                                // type = 2 ("image")

        // D# group 1 (256b): 2D tensor, data_size=4B.
        //   tensor_dim0=128, tensor_dim1=256, tile_dim0=128, tile_dim1=256,
        //   tensor_dim0_stride=128. No padding, no multicast, no barrier.
        v8i_t g1;
        g1[0] = (2 << 16);            // workgroup_mask=0 | data_size=2 (4B)
        g1[1] = (RPQ_CB_DIM << 16);   // atomic_barrier_addr=0 | tensor_dim0.lo16
        g1[2] = (RPQ_CODES << 16);    // tensor_dim0.hi16=0   | tensor_dim1.lo16
        g1[3] = (RPQ_CB_DIM << 16);   // tensor_dim1.hi16=0   | tile_dim0=128
        g1[4] = RPQ_CODES;            // tile_dim1=256        | tile_dim2=0
        g1[5] = RPQ_CB_DIM;           // tensor_dim0_stride.lo32 = 128
        g1[6] = 0;                    // stride0.hi16=0 | tensor_dim1_stride.lo16=0
        g1[7] = 0;

        v4i_t g2 = {0, 0, 0, 0};      // dims 2-4 unused (zeros => size-1 loops)
        v4i_t g3 = {0, 0, 0, 0};

#if defined(__clang_major__) && (__clang_major__ >= 23)
        v8i_t g_extra = {0, 0, 0, 0, 0, 0, 0, 0};
        __builtin_amdgcn_tensor_load_to_lds(g0, g1, g2, g3, g_extra, 0);
#else
        __builtin_amdgcn_tensor_load_to_lds(g0, g1, g2, g3, 0);
#endif
        __builtin_amdgcn_s_wait_tensorcnt(0);  // wave 0 observes DMA done
    }
#else
    // ---- Fallback: cooperative b128 copy of the codebook into LDS ----
    {
        const float4* __restrict__ src =
            (const float4*)(codebooks + (size_t)h * RPQ_CB_ELEMS);
        float4* dst = (float4*)lds;
        for (int i = tid; i < RPQ_CB_ELEMS / 4; i += blockDim.x)
            dst[i] = src[i];
    }
#endif
    __syncthreads();   // release all 8 waves once LDS holds the codebook

    // ---- Gather + stream: one token per wave per iteration ----
    const int wave = tid >> 5;        // wave32
    const int lane = tid & 31;
    const int base = blockIdx.x * RPQ_TOK_PER_BLK;
    const int* __restrict__ idx_h = indices + (size_t)h * num_emb;

#pragma unroll
    for (int it = 0; it < RPQ_TOK_PER_BLK / 8; ++it) {
        const int tok = base + it * 8 + wave;
        if (tok < num_tokens) {
            const int tokid = input[tok];          // broadcast load (uniform in wave)
            const int code  = idx_h[tokid];        // uniform in wave
            // 512B row: lane covers 16B -> ds_load_b128 + global_store_b128
            const float4 v =
                *(const float4*)(lds + code * RPQ_CB_DIM + lane * 4);
            *(float4*)(out + (size_t)tok * RPQ_EMBED + h * RPQ_CB_DIM + lane * 4) = v;
        }
    }
}

extern "C" void kernel_launch(void* const* d_in, const int* in_sizes, int n_in,
                              void* d_out, int out_size, void* d_ws, size_t ws_size,
                              hipStream_t stream) {
    const int*   input     = (const int*)d_in[0];    // (4, 4096) int32
    const int*   indices   = (const int*)d_in[1];    // (16, 50257) int32
    const float* codebooks = (const float*)d_in[2];  // (16, 256, 128) fp32
    float*       out       = (float*)d_out;          // (4, 4096, 2048) fp32

    const int num_tokens = in_sizes[0];              // 16384
    const int num_emb    = in_sizes[1] / RPQ_NUM_CB; // 50257

    // 128 KB dynamic LDS per block (2 blocks still fit in the 320 KB WGP LDS).
    const int shmem_bytes = RPQ_CB_ELEMS * (int)sizeof(float);
    (void)hipFuncSetAttribute((const void*)rpq_embed_kernel,
                              hipFuncAttributeMaxDynamicSharedMemorySize,
                              shmem_bytes);

    dim3 grid((num_tokens + RPQ_TOK_PER_BLK - 1) / RPQ_TOK_PER_BLK, RPQ_NUM_CB);
    dim3 block(256);
    rpq_embed_kernel<<<grid, block, shmem_bytes, stream>>>(
        input, indices, codebooks, out, num_tokens, num_emb);
}